// Decoder_39951785787490
// MI455X (gfx1250) — compile-verified
//
#include <hip/hip_runtime.h>
#include <hip/hip_bf16.h>

// Shapes from the reference
#define N_SPECIES 4
#define M_ROWS    512      // only rows indexed by batch_idx (0..511) matter
#define K1        512      // 2*N_LATENT
#define K2        1024     // H
#define NDIM      1024     // H
#define N_GENES   20000
#define M_PER_SP  4096     // NNZ / N_SPECIES
#define NLAT      256

typedef __attribute__((ext_vector_type(2))) float v2f;
typedef __attribute__((ext_vector_type(8))) float v8f;

// ---------------------------------------------------------------------------
// Fused GEMM + bias + ReLU using V_WMMA_F32_16X16X4_F32 (fp32 matrix core).
//   Hout[s][m][n] = relu( sum_k X[s][m][k] * W[s][k][n] + bias[s][n] )
// layer==0: X built on the fly: cols 0..255 from z[s][m], cols 256..511 from
//           global_latent[batch_idx[s*4096 + m]].
// layer==1: X = Hin[s][m][0..K-1].
// Grid: (NDIM/256, M_ROWS/16, N_SPECIES), block = 256 (8 waves).
// Each wave owns a 16x32 output tile -> TWO independent WMMA accumulation
// chains (hides matrix-pipe accumulate latency); one A fragment feeds both.
// A-tile staged through LDS per 64-wide K-chunk, shared by all 8 waves.
// ---------------------------------------------------------------------------
__global__ __launch_bounds__(256)
void gemm_relu_wmma(const float* __restrict__ z,
                    const float* __restrict__ gl,
                    const int*   __restrict__ bidx,
                    const float* __restrict__ Hin,
                    const float* __restrict__ W,     // [S][K][NDIM]
                    const float* __restrict__ bias,  // [S][NDIM]
                    float*       __restrict__ Hout,  // [S][M_ROWS][NDIM]
                    int K, int layer)
{
    const int s     = blockIdx.z;
    const int tileM = blockIdx.y;            // 16-row tile
    const int tid   = threadIdx.x;
    const int wave  = tid >> 5;              // 0..7
    const int lane  = tid & 31;              // wave32
    const int n0    = blockIdx.x * 256 + wave * 32;   // wave owns 32 columns

    // LDS A-tile: 16 rows x 64 K-columns, padded to kill bank conflicts
    __shared__ float Xs[16][65];

    // staging assignment: each thread loads one float4 of the 16x64 tile
    const int srow  = tid >> 4;              // 0..15
    const int scol4 = (tid & 15) * 4;        // 0,4,...,60
    const int gmrow = tileM * 16 + srow;     // global M row (0..511)

    const int rowL = lane & 15;              // shared by A and B frag layouts
    const int kofs = (lane >> 4) * 2;        // lanes 16..31 hold K+2,K+3

    v8f acc0 = {};
    v8f acc1 = {};

    for (int kk = 0; kk < K; kk += 64) {
        // ---- stage X[16][64] chunk into LDS ----
        float4 xv;
        if (layer == 0) {
            int j = kk + scol4;
            if (j < NLAT) {
                const float* src = z + ((size_t)s * M_PER_SP + gmrow) * NLAT + j;
                xv = *(const float4*)src;
            } else {
                int cell = bidx[(size_t)s * M_PER_SP + gmrow];
                const float* src = gl + (size_t)cell * NLAT + (j - NLAT);
                xv = *(const float4*)src;
            }
        } else {
            const float* src = Hin + ((size_t)s * M_ROWS + gmrow) * (size_t)K + kk + scol4;
            xv = *(const float4*)src;
        }
        Xs[srow][scol4 + 0] = xv.x;
        Xs[srow][scol4 + 1] = xv.y;
        Xs[srow][scol4 + 2] = xv.z;
        Xs[srow][scol4 + 3] = xv.w;
        __syncthreads();

        // ---- 16 WMMA steps over this K-chunk, 2 chains ----
        const float* Wb = W + ((size_t)s * K + kk) * NDIM + (n0 + rowL);
        #pragma unroll
        for (int k = 0; k < 64; k += 4) {
            v2f a, b0, b1;
            // A 16x4 frag: lanes 0-15 -> K=k,k+1 ; lanes 16-31 -> K=k+2,k+3
            a.x = Xs[rowL][k + kofs];
            a.y = Xs[rowL][k + kofs + 1];
            // B 4x16 frags (adjacent 16-col groups; 128B/row across the wave)
            const float* wp0 = Wb + (size_t)(k + kofs) * NDIM;
            const float* wp1 = Wb + (size_t)(k + kofs + 1) * NDIM;
            b0.x = wp0[0];   b0.y = wp1[0];
            b1.x = wp0[16];  b1.y = wp1[16];
            acc0 = __builtin_amdgcn_wmma_f32_16x16x4_f32(
                false, a, false, b0, (short)0, acc0, false, false);
            acc1 = __builtin_amdgcn_wmma_f32_16x16x4_f32(
                false, a, false, b1, (short)0, acc1, false, false);
        }
        __syncthreads();
    }

    // ---- epilogue: bias + ReLU, D layout: VGPR v -> row v (+8 for hi lanes)
    const int c0 = n0 + rowL;
    const float bv0 = bias[(size_t)s * NDIM + c0];
    const float bv1 = bias[(size_t)s * NDIM + c0 + 16];
    const int rbase = tileM * 16 + (lane >> 4) * 8;
    float* outp = Hout + ((size_t)s * M_ROWS) * NDIM + c0;
    #pragma unroll
    for (int v = 0; v < 8; ++v) {
        float v0 = fmaxf(acc0[v] + bv0, 0.0f);
        float v1 = fmaxf(acc1[v] + bv1, 0.0f);
        outp[(size_t)(rbase + v) * NDIM]      = v0;
        outp[(size_t)(rbase + v) * NDIM + 16] = v1;
    }
}

// ---------------------------------------------------------------------------
// Final stage: out[r] = softplus( H2[s][b] . W3[s][:,g] + b3[s][g] )
// One wave per output element; wave32 shuffle reduction.
// ---------------------------------------------------------------------------
__global__ __launch_bounds__(256)
void gather_softplus(const float* __restrict__ H2,   // [S][512][1024]
                     const int*   __restrict__ bidx, // [16384]
                     const int*   __restrict__ gidx, // [16384]
                     const float* __restrict__ W3,   // [S][1024][20000]
                     const float* __restrict__ b3,   // [S][20000]
                     float*       __restrict__ out)  // [16384]
{
    const int wave = threadIdx.x >> 5;
    const int lane = threadIdx.x & 31;
    const int r    = blockIdx.x * 8 + wave;          // output element
    const int s    = r >> 12;                        // r / 4096
    const int b    = bidx[r];
    const int g    = gidx[r];

    const float* hrow = H2 + ((size_t)s * M_ROWS + b) * K2;
    const float* wcol = W3 + (size_t)s * K2 * N_GENES + g;

    float acc = 0.0f;
    #pragma unroll 8
    for (int k = lane; k < K2; k += 32)
        acc += hrow[k] * wcol[(size_t)k * N_GENES];

    // wave32 reduction
    #pragma unroll
    for (int off = 16; off > 0; off >>= 1)
        acc += __shfl_down(acc, off, 32);

    if (lane == 0) {
        float x = acc + b3[(size_t)s * N_GENES + g];
        // numerically stable softplus
        float sp = fmaxf(x, 0.0f) + log1pf(expf(-fabsf(x)));
        out[r] = sp;
    }
}

extern "C" void kernel_launch(void* const* d_in, const int* in_sizes, int n_in,
                              void* d_out, int out_size, void* d_ws, size_t ws_size,
                              hipStream_t stream) {
    // setup_inputs order:
    // 0 values(f32,unused) 1 batch_idx(i32) 2 gene_idx(i32) 3 global_latent
    // 4 z  5 W1  6 b1  7 W2  8 b2  9 W3  10 b3
    const int*   bidx = (const int*)  d_in[1];
    const int*   gidx = (const int*)  d_in[2];
    const float* gl   = (const float*)d_in[3];
    const float* z    = (const float*)d_in[4];
    const float* W1   = (const float*)d_in[5];
    const float* b1   = (const float*)d_in[6];
    const float* W2   = (const float*)d_in[7];
    const float* b2   = (const float*)d_in[8];
    const float* W3   = (const float*)d_in[9];
    const float* b3   = (const float*)d_in[10];

    float* H1 = (float*)d_ws;                                    // 4*512*1024 f32 = 8 MB
    float* H2 = H1 + (size_t)N_SPECIES * M_ROWS * NDIM;          // 8 MB more

    dim3 gdim(NDIM / 256, M_ROWS / 16, N_SPECIES);               // (4, 32, 4)
    gemm_relu_wmma<<<gdim, 256, 0, stream>>>(z, gl, bidx, nullptr, W1, b1, H1, K1, 0);
    gemm_relu_wmma<<<gdim, 256, 0, stream>>>(nullptr, nullptr, nullptr, H1, W2, b2, H2, K2, 1);

    gather_softplus<<<16384 / 8, 256, 0, stream>>>(H2, bidx, gidx, W3, b3, (float*)d_out);
}